// PolicyNetwork1_32933809226463
// MI455X (gfx1250) — compile-verified
//
#include <hip/hip_runtime.h>
#include <math.h>
#include <stdint.h>

typedef __attribute__((ext_vector_type(2))) float v2f;
typedef __attribute__((ext_vector_type(8))) float v8f;
typedef __attribute__((ext_vector_type(4))) unsigned int u32x4;
typedef __attribute__((ext_vector_type(8))) int i32x8;
typedef __attribute__((ext_vector_type(4))) int i32x4;

#define NTHREADS 256

// ---------------------------------------------------------------------------
// Kernel 1: LayerNorm (torch-style unbiased var, ddof=1).
// One block per (b,s) row of D=1024; 256 threads, 4 elements/thread in regs.
// ---------------------------------------------------------------------------
__global__ __launch_bounds__(NTHREADS) void layernorm_kernel(
    const float* __restrict__ ctx, const float* __restrict__ ln_a,
    const float* __restrict__ ln_b, float* __restrict__ x, int D)
{
    __shared__ float red[NTHREADS];
    const size_t row = blockIdx.x;
    const float* c = ctx + row * (size_t)D;
    float*      xo = x   + row * (size_t)D;

    float v[4];
    float s = 0.f;
#pragma unroll
    for (int i = 0; i < 4; ++i) {
        v[i] = c[threadIdx.x + i * NTHREADS];
        s += v[i];
    }
    red[threadIdx.x] = s;
    __syncthreads();
    for (int off = NTHREADS / 2; off > 0; off >>= 1) {
        if (threadIdx.x < off) red[threadIdx.x] += red[threadIdx.x + off];
        __syncthreads();
    }
    const float mean = red[0] / (float)D;
    __syncthreads();

    float ss = 0.f;
#pragma unroll
    for (int i = 0; i < 4; ++i) {
        float d = v[i] - mean;
        ss += d * d;
    }
    red[threadIdx.x] = ss;
    __syncthreads();
    for (int off = NTHREADS / 2; off > 0; off >>= 1) {
        if (threadIdx.x < off) red[threadIdx.x] += red[threadIdx.x + off];
        __syncthreads();
    }
    const float var = red[0] / (float)(D - 1);
    const float inv = 1.0f / (sqrtf(var) + 1e-6f);

#pragma unroll
    for (int i = 0; i < 4; ++i) {
        int d = threadIdx.x + i * NTHREADS;
        xo[d] = ln_a[d] * (v[i] - mean) * inv + ln_b[d];
    }
}

// ---------------------------------------------------------------------------
// TDM helper: issue TENSOR_LOAD_TO_LDS of a 2-D f32 tile (tile_dim0 K-elems x
// tile_dim1 rows) from a row-major tensor with row stride `stride0` elements.
// LDS destination gets 1 dword of padding after every 64 dwords (row stride
// becomes 65 dwords -> conflict-free ds reads of column fragments).
// D# packing per CDNA5 ISA.
// This toolchain exposes the 6-arg builtin: (g0, g1, g2, g3, g_extra, cpol).
// ---------------------------------------------------------------------------
#define KC     64          // K elements per chunk (= 64 dwords per row)
#define LDS_RS 65          // LDS row stride in dwords after TDM padding
#define TILE_M 64          // rows per block tile

__device__ __forceinline__ void tdm_load_x_tile(
    uint64_t gbyte_addr,      // global byte address of tile start
    uint32_t lds_byte_off,    // LDS byte offset of destination buffer
    int tensor_dim0, int tensor_dim1, int stride0)
{
    u32x4 g0;
    g0.x = 1u;                                        // count=1 (valid), user D#
    g0.y = lds_byte_off;                              // lds_addr
    g0.z = (unsigned)(gbyte_addr & 0xFFFFFFFFu);      // global_addr[31:0]
    g0.w = (unsigned)((gbyte_addr >> 32) & 0x01FFFFFFu) | (2u << 30); // addr[56:32] | type=2

    i32x8 g1;
    g1[0] = (2 << 16)       // data_size = 4 bytes
          | (1 << 20)       // pad_enable
          | (5 << 22)       // pad_interval code 5 -> every 64 dwords
          | (0 << 25);      // pad_amount  code 0 -> 1 dword
    g1[1] = (tensor_dim0 & 0xFFFF) << 16;                              // dim0 lo16
    g1[2] = ((tensor_dim0 >> 16) & 0xFFFF) | ((tensor_dim1 & 0xFFFF) << 16);
    g1[3] = ((tensor_dim1 >> 16) & 0xFFFF) | (KC << 16);               // tile_dim0
    g1[4] = TILE_M;                                                    // tile_dim1 (tile_dim2=0)
    g1[5] = stride0;                                                   // tensor_dim0_stride lo32
    g1[6] = 0;
    g1[7] = 0;

    i32x4 z4 = {0, 0, 0, 0};              // groups 2/3 unused (2-D tensor)
    i32x8 z8 = {0, 0, 0, 0, 0, 0, 0, 0};  // extra group unused
    __builtin_amdgcn_tensor_load_to_lds(g0, g1, z4, z4, z8, 0);
}

// ---------------------------------------------------------------------------
// Kernel 2: q = x @ Wq^T + bq ; k = x @ Wk^T + bk via V_WMMA_F32_16X16X4_F32.
//
// Block = 8 waves, block tile 64 rows x 128 cols. Each wave owns a 16-col
// strip and 4 row-tiles for BOTH q and k -> 8 v8f accumulators; per K-step
// it issues 8 WMMAs against 2 global float2 loads (Wq/Wk fragments) and
// 4 LDS float2 loads (A fragments).
//
// The x tile (64 rows x 64 K) is staged into LDS by the Tensor Data Mover,
// double-buffered: wave 0 issues tensor_load_to_lds for chunk c+1 while all
// waves compute on chunk c, then s_wait_tensorcnt(0) + workgroup barrier.
//
// Fragment layouts (CDNA5 ISA, f32 16x16x4, lane=(half,lidx)):
//   A: float2 x[row0 + lidx, k + 2*half]
//   B (= W^T): float2 W[col0 + lidx, k + 2*half]
//   C/D: VGPR v -> element (row0 + v + 8*half, col0 + lidx)
// ---------------------------------------------------------------------------
__global__ __launch_bounds__(NTHREADS) void qk_gemm_wmma_kernel(
    const float* __restrict__ x,   // [R, D]
    const float* __restrict__ Wq,  // [D, D] row-major
    const float* __restrict__ Wk,  // [D, D]
    const float* __restrict__ bq,  // [D]
    const float* __restrict__ bk,  // [D]
    float* __restrict__ q,         // [R, D]
    float* __restrict__ k,         // [R, D]
    int R, int D)
{
    __shared__ float smem[2 * TILE_M * LDS_RS];      // 33,280 B double buffer

    const int wave = threadIdx.x >> 5;
    const int lane = threadIdx.x & 31;
    const int half = lane >> 4;       // 0 or 1
    const int lidx = lane & 15;
    const int col0 = blockIdx.x * 128 + wave * 16;
    const int row0 = blockIdx.y * TILE_M;
    const int ko   = half * 2;        // K sub-offset per lane group

    const float* __restrict__ wqrow = Wq + (size_t)(col0 + lidx) * D + ko;
    const float* __restrict__ wkrow = Wk + (size_t)(col0 + lidx) * D + ko;

    // Flat-address low 32 bits of an LDS pointer == LDS byte offset.
    const uint32_t lds_base = (uint32_t)(uintptr_t)&smem[0];
    const uint64_t gx_base  = (uint64_t)(uintptr_t)x
                            + ((uint64_t)row0 * (uint64_t)D) * 4ull;

    v8f cq[4] = {};
    v8f ck[4] = {};

    const int NCHUNK = D / KC;

    // Prologue: stage chunk 0 into buffer 0.
    if (wave == 0) {
        tdm_load_x_tile(gx_base, lds_base, D, R, D);
        __builtin_amdgcn_s_wait_tensorcnt(0);
    }
    __syncthreads();

    for (int c = 0; c < NCHUNK; ++c) {
        // Overlap: DMA chunk c+1 into the other buffer while computing on c.
        if (wave == 0 && c + 1 < NCHUNK) {
            tdm_load_x_tile(gx_base + (uint64_t)(c + 1) * KC * 4ull,
                            lds_base + ((c + 1) & 1) * (TILE_M * LDS_RS * 4),
                            D, R, D);
        }

        const float* __restrict__ abuf =
            smem + (c & 1) * (TILE_M * LDS_RS) + ko;
        const float* __restrict__ wq_c = wqrow + c * KC;
        const float* __restrict__ wk_c = wkrow + c * KC;

#pragma unroll 4
        for (int kk = 0; kk < KC; kk += 4) {
            v2f bq_f = *(const v2f*)(wq_c + kk);
            v2f bk_f = *(const v2f*)(wk_c + kk);
#pragma unroll
            for (int mt = 0; mt < 4; ++mt) {
                v2f a = *(const v2f*)(abuf + (mt * 16 + lidx) * LDS_RS + kk);
                cq[mt] = __builtin_amdgcn_wmma_f32_16x16x4_f32(
                             false, a, false, bq_f, (short)0, cq[mt], false, false);
                ck[mt] = __builtin_amdgcn_wmma_f32_16x16x4_f32(
                             false, a, false, bk_f, (short)0, ck[mt], false, false);
            }
        }

        if (wave == 0) __builtin_amdgcn_s_wait_tensorcnt(0);
        __syncthreads();   // tile c+1 published; everyone done reading tile c
    }

    const float bqv = bq[col0 + lidx];
    const float bkv = bk[col0 + lidx];
#pragma unroll
    for (int mt = 0; mt < 4; ++mt) {
#pragma unroll
        for (int v = 0; v < 8; ++v) {
            const size_t row = (size_t)(row0 + mt * 16 + v + 8 * half);
            q[row * D + col0 + lidx] = cq[mt][v] + bqv;
            k[row * D + col0 + lidx] = ck[mt][v] + bkv;
        }
    }
}

// ---------------------------------------------------------------------------
// Kernel 3: banded scores + masked 2-way softmax -> per-row (pm, pp, deg).
// One wave per (b,s) row: dot(q[s], k[s±1]) over D, wave32 shuffle reduce.
//   deg=1 : both band entries masked -> softmax row is uniform 1/S.
// ---------------------------------------------------------------------------
__global__ __launch_bounds__(NTHREADS) void band_softmax_kernel(
    const float* __restrict__ q, const float* __restrict__ k,
    const int* __restrict__ eos,
    float* __restrict__ pm, float* __restrict__ pp, float* __restrict__ dg,
    int S, int D)
{
    const int wave = threadIdx.x >> 5;
    const int lane = threadIdx.x & 31;
    const int row  = blockIdx.x * 8 + wave;     // global row in [0, B*S)
    const int b = row / S;
    const int s = row - b * S;

    const float* qs = q + (size_t)row * D;
    float up = 0.f, dn = 0.f;
    if (s + 1 < S) {
        const float* kn = k + (size_t)(row + 1) * D;
        for (int d = lane; d < D; d += 32) up += qs[d] * kn[d];
    }
    if (s > 0) {
        const float* kp = k + (size_t)(row - 1) * D;
        for (int d = lane; d < D; d += 32) dn += qs[d] * kp[d];
    }
#pragma unroll
    for (int off = 16; off > 0; off >>= 1) {
        up += __shfl_xor(up, off, 32);
        dn += __shfl_xor(dn, off, 32);
    }

    if (lane == 0) {
        const float invD = 1.0f / (float)D;
        const float v1 = up * invD;   // score(s, s+1)
        const float v0 = dn * invD;   // score(s, s-1)
        const size_t rbase = ((size_t)b * S + s) * S;
        const int m1 = (s + 1 < S) ? (eos[rbase + s + 1] & 1) : 0;
        const int m0 = (s > 0)     ? (eos[rbase + s - 1] & 1) : 0;
        const float invS = 1.0f / (float)S;
        float Pm, Pp, Dg;
        if (!m0 && !m1) {               // all -1e9 -> uniform softmax
            Pm = invS; Pp = invS; Dg = 1.f;
        } else if (m0 && m1) {          // 2-way softmax, rest underflow to 0
            float z  = fmaxf(v0, v1);
            float e0 = expf(v0 - z);
            float e1 = expf(v1 - z);
            float r  = 1.0f / (e0 + e1);
            Pm = e0 * r; Pp = e1 * r; Dg = 0.f;
        } else if (m1) { Pm = 0.f; Pp = 1.f; Dg = 0.f; }
        else           { Pm = 1.f; Pp = 0.f; Dg = 0.f; }
        pm[row] = Pm; pp[row] = Pp; dg[row] = Dg;
    }
}

// ---------------------------------------------------------------------------
// Kernel 4: per-batch prefix sum of l[s] = log(neibor[s,s+1] + 1e-9), in f64
// (cum reaches ~-2e4; f32 prefix differencing would wipe out small deltas).
// ---------------------------------------------------------------------------
__global__ void scan_kernel(
    const float* __restrict__ pm, const float* __restrict__ pp,
    const float* __restrict__ dg, double* __restrict__ cum, int S)
{
    if (threadIdx.x != 0) return;
    const int base = blockIdx.x * S;
    const float invS = 1.0f / (float)S;
    double c = 0.0;
    cum[base] = 0.0;
    for (int s = 0; s + 1 < S; ++s) {
        float p_up = (dg[base + s]     > 0.5f) ? invS : pp[base + s];     // p(s, s+1)
        float p_dn = (dg[base + s + 1] > 0.5f) ? invS : pm[base + s + 1]; // p(s+1, s)
        float neib = sqrtf(p_up * p_dn + 1e-9f);
        c += (double)logf(neib + 1e-9f);
        cum[base + s + 1] = c;
    }
}

// ---------------------------------------------------------------------------
// Kernel 5: write the two [B,S,S] outputs (memory-bound, one pass).
//   neibor[s,t] = sqrt(p(s,t)*p(t,s) + 1e-9)
//   g[s,t] = (s==t) ? sqrt(p(s,s)^2+1e-9) : exp(cum[hi]-cum[lo]) + 1e-9
// ---------------------------------------------------------------------------
__global__ __launch_bounds__(NTHREADS) void output_kernel(
    const float* __restrict__ pm, const float* __restrict__ pp,
    const float* __restrict__ dg, const double* __restrict__ cum,
    float* __restrict__ g, float* __restrict__ neib, int S)
{
    const int row  = blockIdx.x;        // b*S + s
    const int b    = row / S;
    const int s    = row - b * S;
    const int base = b * S;
    const float invS = 1.0f / (float)S;

    const bool  degS = dg[row] > 0.5f;
    const float pmS  = pm[row];
    const float ppS  = pp[row];
    const double cS  = cum[row];
    const float pss  = degS ? invS : 0.f;
    const float gdiag = sqrtf(pss * pss + 1e-9f);

    float* __restrict__ grow = g    + (size_t)row * S;
    float* __restrict__ nrow = neib + (size_t)row * S;

    for (int t = threadIdx.x; t < S; t += NTHREADS) {
        const int rt = base + t;
        const bool degT = dg[rt] > 0.5f;
        const float pst = degS ? invS
                        : (t == s - 1 ? pmS : (t == s + 1 ? ppS : 0.f));
        const float pts = degT ? invS
                        : (s == t - 1 ? pm[rt] : (s == t + 1 ? pp[rt] : 0.f));
        nrow[t] = sqrtf(pst * pts + 1e-9f);

        float gv;
        if (t == s) {
            gv = gdiag;
        } else {
            double d = (t > s) ? (cum[rt] - cS) : (cS - cum[rt]);
            gv = expf((float)d) + 1e-9f;
        }
        grow[t] = gv;
    }
}

// ---------------------------------------------------------------------------
extern "C" void kernel_launch(void* const* d_in, const int* in_sizes, int n_in,
                              void* d_out, int out_size, void* d_ws, size_t ws_size,
                              hipStream_t stream)
{
    const int B = 8, S = 1024, D = 1024;
    const size_t R = (size_t)B * S;               // 8192 rows

    const float* context = (const float*)d_in[0];
    const int*   eos     = (const int*)  d_in[1];
    const float* Wq      = (const float*)d_in[2];
    const float* bq      = (const float*)d_in[3];
    const float* Wk      = (const float*)d_in[4];
    const float* bk      = (const float*)d_in[5];
    const float* ln_a    = (const float*)d_in[6];
    const float* ln_b    = (const float*)d_in[7];

    // Workspace layout (all offsets 8-byte aligned):
    float*  x   = (float*)d_ws;          // R*D
    float*  q   = x + R * D;             // R*D
    float*  k   = q + R * D;             // R*D
    float*  pm  = k + R * D;             // R
    float*  pp  = pm + R;                // R
    float*  dg  = pp + R;                // R
    double* cum = (double*)(dg + R);     // R doubles

    layernorm_kernel<<<(unsigned)R, NTHREADS, 0, stream>>>(context, ln_a, ln_b, x, D);

    dim3 gemm_grid(D / 128, (unsigned)(R / TILE_M));
    qk_gemm_wmma_kernel<<<gemm_grid, NTHREADS, 0, stream>>>(x, Wq, Wk, bq, bk, q, k,
                                                            (int)R, D);

    band_softmax_kernel<<<(unsigned)(R / 8), NTHREADS, 0, stream>>>(q, k, eos, pm, pp, dg, S, D);

    scan_kernel<<<B, 32, 0, stream>>>(pm, pp, dg, cum, S);

    float* gout = (float*)d_out;
    float* nout = gout + (size_t)B * S * S;
    output_kernel<<<(unsigned)R, NTHREADS, 0, stream>>>(pm, pp, dg, cum, gout, nout, S);
}